// GEVBeamformer_81698867905094
// MI455X (gfx1250) — compile-verified
//
#include <hip/hip_runtime.h>
#include <hip/hip_bf16.h>
#include <math.h>

#define B_ 4
#define M_ 6
#define F_ 513
#define T_ 2000
#define EPS_ 1e-6f

typedef float v2f __attribute__((ext_vector_type(2)));
typedef float v8f __attribute__((ext_vector_type(8)));

// ---------------------------------------------------------------------------
// Phase 1: per-(b,f) 6x6 generalized eigensolve. One thread per problem.
// Writes normalized max-eigenvector (beamforming weights) to ws: [B*F][6].
// ---------------------------------------------------------------------------
__global__ __launch_bounds__(64) void gev_solve_kernel(const float* __restrict__ tgt,
                                                       const float* __restrict__ noi,
                                                       float* __restrict__ bfw) {
    int gid = blockIdx.x * blockDim.x + threadIdx.x;
    if (gid >= B_ * F_) return;
    int b = gid / F_;
    int f = gid - b * F_;
    // scm layout: (B, M, M, F) -> idx = ((b*6+i)*6+j)*F + f
    int base = (b * 36) * F_ + f;

    float a[6][6], n[6][6];
    for (int i = 0; i < 6; ++i)
        for (int j = 0; j < 6; ++j) {
            int idx = base + (i * 6 + j) * F_;
            a[i][j] = tgt[idx];
            n[i][j] = noi[idx];
        }

    // condition_scm: (n + eps*tr/6 * I) / (1+eps)
    float tr = 0.f;
    for (int i = 0; i < 6; ++i) tr += n[i][i];
    float sc = EPS_ * tr / 6.f;
    float inv1pe = 1.f / (1.f + EPS_);
    for (int i = 0; i < 6; ++i)
        for (int j = 0; j < 6; ++j)
            n[i][j] = (n[i][j] + ((i == j) ? sc : 0.f)) * inv1pe;

    // Cholesky: n = L L^T (lower)
    float L[6][6];
    for (int i = 0; i < 6; ++i)
        for (int j = 0; j < 6; ++j) L[i][j] = 0.f;
    for (int k = 0; k < 6; ++k) {
        float d = n[k][k];
        for (int j = 0; j < k; ++j) d -= L[k][j] * L[k][j];
        d = fmaxf(d, 1e-30f);
        float lkk = sqrtf(d);
        L[k][k] = lkk;
        float r = 1.f / lkk;
        for (int i = k + 1; i < 6; ++i) {
            float s = n[i][k];
            for (int j = 0; j < k; ++j) s -= L[i][j] * L[k][j];
            L[i][k] = s * r;
        }
    }

    // Li = inverse(L), lower triangular (forward substitution per column)
    float Li[6][6];
    for (int i = 0; i < 6; ++i)
        for (int j = 0; j < 6; ++j) Li[i][j] = 0.f;
    for (int j = 0; j < 6; ++j) {
        Li[j][j] = 1.f / L[j][j];
        for (int i = j + 1; i < 6; ++i) {
            float s = 0.f;
            for (int k = j; k < i; ++k) s += L[i][k] * Li[k][j];
            Li[i][j] = -s / L[i][i];
        }
    }

    // cm = Li * a * Li^T  (symmetric)
    float tmp[6][6];
    for (int i = 0; i < 6; ++i)
        for (int j = 0; j < 6; ++j) {
            float s = 0.f;
            for (int k = 0; k <= i; ++k) s += Li[i][k] * a[k][j];
            tmp[i][j] = s;
        }
    float cm[6][6];
    for (int i = 0; i < 6; ++i)
        for (int j = 0; j < 6; ++j) {
            float s = 0.f;
            for (int k = 0; k <= j; ++k) s += tmp[i][k] * Li[j][k];
            cm[i][j] = s;
        }
    // symmetrize (numerical)
    for (int i = 0; i < 6; ++i)
        for (int j = i + 1; j < 6; ++j) {
            float s = 0.5f * (cm[i][j] + cm[j][i]);
            cm[i][j] = s;
            cm[j][i] = s;
        }

    // Cyclic Jacobi eigensolver with eigenvector accumulation
    float V[6][6];
    for (int i = 0; i < 6; ++i)
        for (int j = 0; j < 6; ++j) V[i][j] = (i == j) ? 1.f : 0.f;
    for (int sweep = 0; sweep < 12; ++sweep) {
        for (int p = 0; p < 5; ++p) {
            for (int q = p + 1; q < 6; ++q) {
                float apq = cm[p][q];
                if (fabsf(apq) > 1e-32f) {
                    float theta = (cm[q][q] - cm[p][p]) / (2.f * apq);
                    float tn = ((theta >= 0.f) ? 1.f : -1.f) /
                               (fabsf(theta) + sqrtf(theta * theta + 1.f));
                    float c = 1.f / sqrtf(tn * tn + 1.f);
                    float s = tn * c;
                    // A' = J^T A J : columns p,q then rows p,q
                    for (int k = 0; k < 6; ++k) {
                        float akp = cm[k][p], akq = cm[k][q];
                        cm[k][p] = c * akp - s * akq;
                        cm[k][q] = s * akp + c * akq;
                    }
                    for (int k = 0; k < 6; ++k) {
                        float apk = cm[p][k], aqk = cm[q][k];
                        cm[p][k] = c * apk - s * aqk;
                        cm[q][k] = s * apk + c * aqk;
                    }
                    for (int k = 0; k < 6; ++k) {
                        float vkp = V[k][p], vkq = V[k][q];
                        V[k][p] = c * vkp - s * vkq;
                        V[k][q] = s * vkp + c * vkq;
                    }
                }
            }
        }
    }

    // pick max-eigenvalue column
    int kmax = 0;
    float lmax = cm[0][0];
    for (int k = 1; k < 6; ++k) {
        if (cm[k][k] > lmax) { lmax = cm[k][k]; kmax = k; }
    }
    float v[6];
    for (int i = 0; i < 6; ++i) v[i] = V[i][kmax];

    // evec = Li^T * v  ; e[i] = sum_{j>=i} Li[j][i] * v[j]
    float e[6];
    float nrm2 = 0.f;
    for (int i = 0; i < 6; ++i) {
        float s = 0.f;
        for (int j = i; j < 6; ++j) s += Li[j][i] * v[j];
        e[i] = s;
        nrm2 += s * s;
    }
    float rn = 1.f / sqrtf(fmaxf(nrm2, 1e-30f));
    for (int m = 0; m < 6; ++m) bfw[gid * 6 + m] = e[m] * rn;
}

// ---------------------------------------------------------------------------
// Phase 2: out[b,f,t] = sum_m bf[b,f,m] * mix[b,m,f,t]  via V_WMMA_F32_16X16X4
// One block (8 waves) per (b,f); each wave handles 32 t's per iteration
// using two 16-t WMMA tiles; full-wave coalesced 128B stores.
// A rows broadcast bf -> D rows all equal out[t-tile]; A and B use the
// identical (lane-half, vgpr)->K placement so the K-contraction is invariant
// to the hardware's internal K ordering.
// ---------------------------------------------------------------------------
__global__ __launch_bounds__(256) void bf_apply_kernel(const float* __restrict__ mix,
                                                       const float* __restrict__ bfw,
                                                       float* __restrict__ out) {
    const int bf_id = blockIdx.x;             // 0 .. B*F-1
    const int b = bf_id / F_;
    const int f = bf_id - b * F_;
    const int lane = threadIdx.x & 31;
    const int wave = threadIdx.x >> 5;        // 0..7 (uniform per wave)
    const bool lo = lane < 16;
    const int ln = lane & 15;

    // beamforming weights (uniform across wave -> broadcast loads)
    const float* w = bfw + bf_id * 6;
    const float w0 = w[0], w1 = w[1], w2 = w[2], w3 = w[3], w4 = w[4], w5 = w[5];

    // A operand: every row i holds bf[k].  K placement: lanes<16 -> {K0,K1} in
    // {vgpr0,vgpr1}; lanes>=16 -> {K2,K3}.
    v2f A1, A2;
    A1.x = lo ? w0 : w2;
    A1.y = lo ? w1 : w3;
    A2.x = lo ? w4 : 0.f;   // second WMMA: K0=m4, K1=m5, K2=K3=0 (pad)
    A2.y = lo ? w5 : 0.f;

    // mix row pointers: mix[b][m][f][:] ; row stride = F*T
    const size_t rs = (size_t)F_ * T_;
    const float* base = mix + (size_t)(b * 6) * rs + (size_t)f * T_;
    const int rm = lo ? 0 : 2;                        // rows for first WMMA B
    const float* rowA = base + (size_t)rm * rs;       // m = rm
    const float* rowB = base + (size_t)(rm + 1) * rs; // m = rm+1
    const float* row4 = base + (size_t)4 * rs;
    const float* row5 = base + (size_t)5 * rs;

    float* outp = out + (size_t)bf_id * T_;

    const int nchunk = (T_ + 31) / 32;  // 63
    for (int c = wave; c < nchunk; c += 8) {
        const int t0 = c * 32;
        int tA = t0 + ln;
        int tB = t0 + 16 + ln;
        int tAc = (tA < T_) ? tA : (T_ - 1);
        int tBc = (tB < T_) ? tB : (T_ - 1);

        // prefetch next chunk for this wave (+8*32 floats = +1KB)
        __builtin_prefetch(rowA + tAc + 256, 0, 1);
        __builtin_prefetch(rowB + tAc + 256, 0, 1);

        // B operands, tile A (t0..t0+15) and tile B (t0+16..t0+31)
        v2f B1a, B2a, B1b, B2b;
        B1a.x = rowA[tAc];
        B1a.y = rowB[tAc];
        B2a.x = row4[tAc];   // hi-lane contents are don't-care: A2 K2,K3 == 0
        B2a.y = row5[tAc];
        B1b.x = rowA[tBc];
        B1b.y = rowB[tBc];
        B2b.x = row4[tBc];
        B2b.y = row5[tBc];

        v8f accA = {0.f, 0.f, 0.f, 0.f, 0.f, 0.f, 0.f, 0.f};
        v8f accB = {0.f, 0.f, 0.f, 0.f, 0.f, 0.f, 0.f, 0.f};
        accA = __builtin_amdgcn_wmma_f32_16x16x4_f32(false, A1, false, B1a,
                                                     (short)0, accA, false, false);
        accA = __builtin_amdgcn_wmma_f32_16x16x4_f32(false, A2, false, B2a,
                                                     (short)0, accA, false, false);
        accB = __builtin_amdgcn_wmma_f32_16x16x4_f32(false, A1, false, B1b,
                                                     (short)0, accB, false, false);
        accB = __builtin_amdgcn_wmma_f32_16x16x4_f32(false, A2, false, B2b,
                                                     (short)0, accB, false, false);

        // D vgpr0: lanes 0-15 hold out[t0+N] (tile A), lanes 16-31 hold
        // out[t0+16+N] (tile B) -> one coalesced 32-float store.
        float val = lo ? accA[0] : accB[0];
        int ts = t0 + lane;
        if (ts < T_) outp[ts] = val;
    }
}

extern "C" void kernel_launch(void* const* d_in, const int* in_sizes, int n_in,
                              void* d_out, int out_size, void* d_ws, size_t ws_size,
                              hipStream_t stream) {
    const float* mix = (const float*)d_in[0];        // (B, M, F, T)
    const float* tgt = (const float*)d_in[1];        // (B, M, M, F)
    const float* noi = (const float*)d_in[2];        // (B, M, M, F)
    float* out = (float*)d_out;                      // (B, F, T)
    float* bfw = (float*)d_ws;                       // (B*F, 6) scratch

    const int nprob = B_ * F_;                       // 2052
    gev_solve_kernel<<<(nprob + 63) / 64, 64, 0, stream>>>(tgt, noi, bfw);
    bf_apply_kernel<<<nprob, 256, 0, stream>>>(mix, bfw, out);
}